// GIN_58308476010614
// MI455X (gfx1250) — compile-verified
//
#include <hip/hip_runtime.h>

#define NNODES 100000
#define NEDGES 3200000
#define NGRAPHS 1024
#define HID 64
#define ZP 65  // LDS pitch for 16xK z tiles: odd -> bank-conflict free

typedef __attribute__((ext_vector_type(2))) float v2f;
typedef __attribute__((ext_vector_type(8))) float v8f;

// ---------------------------------------------------------------------------
// Wave-cooperative GEMM via V_WMMA_F32_16X16X4_F32:
//   out[16][64] = relu(Z[16][K] @ W[K][64] + bias)
// Z in LDS at pitch ZP; W row-major [K][64] in LDS; bias[64] in LDS.
// A layout (ISA 7.12.2, 32-bit 16x4): lane l: M=l%16, VGPR j holds K = 2*(l/16)+j
// B layout mirrors with N=l%16. C/D: element (M = v + 8*(l/16), N = nt*16 + l%16).
// ---------------------------------------------------------------------------
template <int K>
__device__ __forceinline__ void wave_gemm_bias_relu(const float* lZ, const float* lW,
                                                    const float* lBias, v8f out[4],
                                                    int lane) {
  const int m  = lane & 15;
  const int kh = lane >> 4;  // which K-pair within each 4-wide K step
  v8f acc[4];
#pragma unroll
  for (int nt = 0; nt < 4; ++nt)
#pragma unroll
    for (int v = 0; v < 8; ++v) acc[nt][v] = 0.0f;

#pragma unroll
  for (int k0 = 0; k0 < K; k0 += 4) {
    const int ka = k0 + 2 * kh;
    v2f a;
    a[0] = lZ[m * ZP + ka];
    a[1] = lZ[m * ZP + ka + 1];
#pragma unroll
    for (int nt = 0; nt < 4; ++nt) {
      v2f b;
      b[0] = lW[ka * HID + nt * 16 + m];
      b[1] = lW[(ka + 1) * HID + nt * 16 + m];
      acc[nt] = __builtin_amdgcn_wmma_f32_16x16x4_f32(false, a, false, b,
                                                      (short)0, acc[nt], false, false);
    }
  }
#pragma unroll
  for (int nt = 0; nt < 4; ++nt) {
    const float bia = lBias[nt * 16 + m];
#pragma unroll
    for (int v = 0; v < 8; ++v) out[nt][v] = fmaxf(acc[nt][v] + bia, 0.0f);
  }
}

// ---------------------------------------------------------------------------
// Utility: zero a float buffer
// ---------------------------------------------------------------------------
__global__ void zero_kernel(float* __restrict__ p, int n) {
  int i = blockIdx.x * blockDim.x + threadIdx.x;
  if (i < n) p[i] = 0.0f;
}

// ---------------------------------------------------------------------------
// Layer-1 edge scatter: agg1[dst][0:5] += x[src][0:5], one thread per edge
// ---------------------------------------------------------------------------
__global__ void edge_agg5(const float* __restrict__ x, const long long* __restrict__ ei,
                          float* __restrict__ agg) {
  int e = blockIdx.x * blockDim.x + threadIdx.x;
  if (e >= NEDGES) return;
  int s = (int)ei[e];
  int d = (int)ei[NEDGES + e];
#pragma unroll
  for (int c = 0; c < 5; ++c) atomicAdd(&agg[d * 5 + c], x[s * 5 + c]);
}

// ---------------------------------------------------------------------------
// Layer-2 edge scatter: thread = (edge, channel); 64 consecutive lanes share an
// edge -> coalesced 256B gather of h[src] and coalesced atomic burst on agg[dst].
// L2-resident (h & agg ~25.6MB each << 192MB L2).
// ---------------------------------------------------------------------------
__global__ void edge_agg64(const float* __restrict__ h, const long long* __restrict__ ei,
                           float* __restrict__ agg) {
  int gid = blockIdx.x * blockDim.x + threadIdx.x;
  int e = gid >> 6;
  int c = gid & 63;
  if (e >= NEDGES) return;
  int s = (int)ei[e];
  int d = (int)ei[NEDGES + e];
  atomicAdd(&agg[d * HID + c], h[s * HID + c]);
}

// ---------------------------------------------------------------------------
// GIN MLP layer 1: h1 = relu( relu((x+agg1) @ W1a + b1a) @ W1b + b1b )
// K=5 first GEMM zero-padded to K=8 so it also runs on WMMA.
// 128 threads = 4 waves; each wave owns a 16-node tile.
// ---------------------------------------------------------------------------
__global__ __launch_bounds__(128) void gin_mlp1(
    const float* __restrict__ x, const float* __restrict__ agg1,
    const float* __restrict__ W1a, const float* __restrict__ b1a,
    const float* __restrict__ W1b, const float* __restrict__ b1b,
    float* __restrict__ h1) {
  __shared__ float lWa[8 * HID];
  __shared__ float lWb[HID * HID];
  __shared__ float lBa[HID];
  __shared__ float lBb[HID];
  __shared__ float zbuf[4 * 16 * ZP];

  const int tid = threadIdx.x;
  for (int i = tid; i < 8 * HID; i += 128) {
    int r = i >> 6, c = i & 63;
    lWa[i] = (r < 5) ? W1a[r * HID + c] : 0.0f;
  }
  for (int i = tid; i < HID * HID; i += 128) lWb[i] = W1b[i];
  if (tid < HID) { lBa[tid] = b1a[tid]; lBb[tid] = b1b[tid]; }
  __syncthreads();

  const int wid = tid >> 5, lane = tid & 31;
  float* lZ = &zbuf[wid * 16 * ZP];
  const int tileBase = (blockIdx.x * 4 + wid) * 16;

  // stage z1 = x + agg1 (16 nodes x 8, cols 5..7 = 0)
  for (int i = lane; i < 16 * 8; i += 32) {
    int mm = i >> 3, k = i & 7;
    int node = tileBase + mm;
    float v = 0.0f;
    if (k < 5 && node < NNODES) v = x[node * 5 + k] + agg1[node * 5 + k];
    lZ[mm * ZP + k] = v;
  }

  v8f t[4];
  wave_gemm_bias_relu<8>(lZ, lWa, lBa, t, lane);

  // bounce intermediate tile through per-wave LDS (in-order DS within a wave)
  const int m = lane & 15, kh = lane >> 4;
#pragma unroll
  for (int nt = 0; nt < 4; ++nt)
#pragma unroll
    for (int v = 0; v < 8; ++v) lZ[(v + 8 * kh) * ZP + nt * 16 + m] = t[nt][v];

  v8f h[4];
  wave_gemm_bias_relu<HID>(lZ, lWb, lBb, h, lane);

#pragma unroll
  for (int v = 0; v < 8; ++v) {
    int node = tileBase + v + 8 * kh;
    if (node < NNODES) {
#pragma unroll
      for (int nt = 0; nt < 4; ++nt) h1[node * HID + nt * 16 + m] = h[nt][v];
    }
  }
}

// ---------------------------------------------------------------------------
// GIN MLP layer 2: h2 = relu( relu((h1+agg2) @ W2a + b2a) @ W2b + b2b )
// ---------------------------------------------------------------------------
__global__ __launch_bounds__(128) void gin_mlp2(
    const float* __restrict__ hin, const float* __restrict__ agg,
    const float* __restrict__ Wa, const float* __restrict__ ba,
    const float* __restrict__ Wb, const float* __restrict__ bb,
    float* __restrict__ hout) {
  __shared__ float lWa[HID * HID];
  __shared__ float lWb[HID * HID];
  __shared__ float lBa[HID];
  __shared__ float lBb[HID];
  __shared__ float zbuf[4 * 16 * ZP];

  const int tid = threadIdx.x;
  for (int i = tid; i < HID * HID; i += 128) {
    lWa[i] = Wa[i];
    lWb[i] = Wb[i];
  }
  if (tid < HID) { lBa[tid] = ba[tid]; lBb[tid] = bb[tid]; }
  __syncthreads();

  const int wid = tid >> 5, lane = tid & 31;
  float* lZ = &zbuf[wid * 16 * ZP];
  const int tileBase = (blockIdx.x * 4 + wid) * 16;

  // stage z2 = h1 + agg2 (16 nodes x 64), coalesced global reads
  for (int i = lane; i < 16 * HID; i += 32) {
    int mm = i >> 6, k = i & 63;
    int node = tileBase + mm;
    float v = 0.0f;
    if (node < NNODES) v = hin[node * HID + k] + agg[node * HID + k];
    lZ[mm * ZP + k] = v;
  }

  v8f t[4];
  wave_gemm_bias_relu<HID>(lZ, lWa, lBa, t, lane);

  const int m = lane & 15, kh = lane >> 4;
#pragma unroll
  for (int nt = 0; nt < 4; ++nt)
#pragma unroll
    for (int v = 0; v < 8; ++v) lZ[(v + 8 * kh) * ZP + nt * 16 + m] = t[nt][v];

  v8f h[4];
  wave_gemm_bias_relu<HID>(lZ, lWb, lBb, h, lane);

#pragma unroll
  for (int v = 0; v < 8; ++v) {
    int node = tileBase + v + 8 * kh;
    if (node < NNODES) {
#pragma unroll
      for (int nt = 0; nt < 4; ++nt) hout[node * HID + nt * 16 + m] = h[nt][v];
    }
  }
}

// ---------------------------------------------------------------------------
// Global mean pool (scatter part): sums[g] += h2[n]; counts[g] += 1
// ---------------------------------------------------------------------------
__global__ void pool_kernel(const float* __restrict__ h2, const long long* __restrict__ batch,
                            float* __restrict__ sums, float* __restrict__ cnts) {
  int gid = blockIdx.x * blockDim.x + threadIdx.x;
  int n = gid >> 6;
  int c = gid & 63;
  if (n >= NNODES) return;
  int g = (int)batch[n];
  atomicAdd(&sums[g * HID + c], h2[n * HID + c]);
  if (c == 0) atomicAdd(&cnts[g], 1.0f);
}

// ---------------------------------------------------------------------------
// Classifier: out[g][o] = (sums[g]/max(cnt,1)) . Wc[:,o] + bc[o]
// ---------------------------------------------------------------------------
__global__ void classify_kernel(const float* __restrict__ sums, const float* __restrict__ cnts,
                                const float* __restrict__ Wc, const float* __restrict__ bc,
                                float* __restrict__ out) {
  int t = blockIdx.x * blockDim.x + threadIdx.x;
  if (t >= NGRAPHS * 2) return;
  int g = t >> 1, o = t & 1;
  float inv = 1.0f / fmaxf(cnts[g], 1.0f);
  float acc = 0.0f;
#pragma unroll
  for (int hh = 0; hh < HID; ++hh) acc += sums[g * HID + hh] * inv * Wc[hh * 2 + o];
  out[t] = acc + bc[o];
}

// ---------------------------------------------------------------------------
extern "C" void kernel_launch(void* const* d_in, const int* in_sizes, int n_in,
                              void* d_out, int out_size, void* d_ws, size_t ws_size,
                              hipStream_t stream) {
  const float* x       = (const float*)d_in[0];
  const long long* ei  = (const long long*)d_in[1];
  const long long* bat = (const long long*)d_in[2];
  const float* W1a = (const float*)d_in[3];
  const float* b1a = (const float*)d_in[4];
  const float* W1b = (const float*)d_in[5];
  const float* b1b = (const float*)d_in[6];
  const float* W2a = (const float*)d_in[7];
  const float* b2a = (const float*)d_in[8];
  const float* W2b = (const float*)d_in[9];
  const float* b2b = (const float*)d_in[10];
  const float* Wc  = (const float*)d_in[11];
  const float* bc  = (const float*)d_in[12];
  float* out = (float*)d_out;

  // workspace layout (floats): agg1 | h1 | agg2 | h2 | sums | counts  (~79 MB)
  float* base = (float*)d_ws;
  float* agg1 = base;                       // 500000
  float* h1   = agg1 + 500000;              // 6400000
  float* agg2 = h1 + NNODES * HID;          // 6400000
  float* h2   = agg2 + NNODES * HID;        // 6400000
  float* sums = h2 + NNODES * HID;          // 65536
  float* cnts = sums + NGRAPHS * HID;       // 1024
  (void)in_sizes; (void)n_in; (void)out_size; (void)ws_size;

  const int TB = 256;

  // ---- layer 1 ----
  zero_kernel<<<(NNODES * 5 + TB - 1) / TB, TB, 0, stream>>>(agg1, NNODES * 5);
  edge_agg5<<<(NEDGES + TB - 1) / TB, TB, 0, stream>>>(x, ei, agg1);
  gin_mlp1<<<(NNODES + 63) / 64, 128, 0, stream>>>(x, agg1, W1a, b1a, W1b, b1b, h1);

  // ---- layer 2 ----
  zero_kernel<<<(NNODES * HID + TB - 1) / TB, TB, 0, stream>>>(agg2, NNODES * HID);
  edge_agg64<<<(NEDGES * 64 + TB - 1) / TB, TB, 0, stream>>>(h1, ei, agg2);
  gin_mlp2<<<(NNODES + 63) / 64, 128, 0, stream>>>(h1, agg2, W2a, b2a, W2b, b2b, h2);

  // ---- mean pool + classifier ----
  zero_kernel<<<(NGRAPHS * (HID + 1) + TB - 1) / TB, TB, 0, stream>>>(sums, NGRAPHS * (HID + 1));
  pool_kernel<<<(NNODES * 64 + TB - 1) / TB, TB, 0, stream>>>(h2, bat, sums, cnts);
  classify_kernel<<<(NGRAPHS * 2 + TB - 1) / TB, TB, 0, stream>>>(sums, cnts, Wc, bc, out);
}